// RowAttentionWithPairBias_80169859547199
// MI455X (gfx1250) — compile-verified
//
#include <hip/hip_runtime.h>
#include <hip/hip_bf16.h>

#define BATCH 2
#define LSEQ 1024
#define DNODE 128
#define DPAIR 64
#define NHEAD 4
#define DHEAD 32
#define JSPLIT 4
#define JTILES (LSEQ / JSPLIT / 16)   // 16 j-tiles of 16 per split

typedef __attribute__((ext_vector_type(2))) float v2f;
typedef __attribute__((ext_vector_type(8))) float v8f;

// A-frag / B-frag for V_WMMA_F32_16X16X4_F32, row-major source:
// row = lane%16, cols = kbase + 2*(lane/16) + {0,1}
__device__ __forceinline__ v2f frag_rm(const float* p, int ld, int lane) {
  const float* q = p + (lane & 15) * ld + ((lane >> 4) << 1);
  v2f r; r.x = q[0]; r.y = q[1]; return r;
}
// B-frag where B[k][n] is stored row-major as src[k*ld + n]
__device__ __forceinline__ v2f frag_cm(const float* p, int ld, int lane) {
  const int n = lane & 15;
  const int k = (lane >> 4) << 1;
  v2f r; r.x = p[k * ld + n]; r.y = p[(k + 1) * ld + n]; return r;
}
__device__ __forceinline__ v8f wmma4(v2f a, v2f b, v8f c) {
  return __builtin_amdgcn_wmma_f32_16x16x4_f32(false, a, false, b, (short)0, c, false, false);
}

// Async global->LDS b128 copies. One shared literal offset advances BOTH the
// global source and LDS destination (CDNA5 ISA).
// Tracked with ASYNCcnt.
template <int I>
__device__ __forceinline__ void axrep(unsigned lds, unsigned long long gsrc) {
  if constexpr (I > 0) {
    axrep<I - 1>(lds, gsrc);
    asm volatile("global_load_async_to_lds_b128 %0, %1, off offset:%2"
                 :: "v"(lds), "v"(gsrc), "n"((I - 1) * 128) : "memory");
  }
}
__device__ __forceinline__ void issue_tile_copy(const float* gbase, const float* lbase) {
  const unsigned lds = (unsigned)(size_t)(const void*)lbase;            // low 32b = LDS offset
  const unsigned long long g = (unsigned long long)(size_t)(const void*)gbase;
  axrep<32>(lds, g);                                                    // 32 x 16B/lane = 512B/lane
}

// ---------------- Kernel 1: LayerNorm + Q/K/V/G projections ----------------
__global__ __launch_bounds__(128) void afp_proj(
    const float* __restrict__ s, const float* __restrict__ lnw, const float* __restrict__ lnb,
    const float* __restrict__ Wq, const float* __restrict__ Wk, const float* __restrict__ Wv,
    const float* __restrict__ Wg, const float* __restrict__ bg,
    float* __restrict__ Qo, float* __restrict__ Ko, float* __restrict__ Vo, float* __restrict__ Go)
{
  __shared__ float zs[16 * DNODE];
  const int row0 = blockIdx.x * 16;
  const int t = threadIdx.x;
  {
    const int r  = t >> 3;
    const int cb = (t & 7) * 16;
    const float* x = s + (size_t)(row0 + r) * DNODE + cb;
    float v[16], sum = 0.f, sq = 0.f;
#pragma unroll
    for (int i = 0; i < 16; ++i) { v[i] = x[i]; sum += v[i]; sq += v[i] * v[i]; }
#pragma unroll
    for (int m = 1; m < 8; m <<= 1) { sum += __shfl_xor(sum, m, 32); sq += __shfl_xor(sq, m, 32); }
    const float mu  = sum * (1.f / 128.f);
    const float var = sq * (1.f / 128.f) - mu * mu;
    const float inv = rsqrtf(var + 1e-5f);
#pragma unroll
    for (int i = 0; i < 16; ++i)
      zs[r * DNODE + cb + i] = (v[i] - mu) * inv * lnw[cb + i] + lnb[cb + i];
  }
  __syncthreads();
  const int wave = t >> 5, lane = t & 31;
  const float* W = (wave == 0) ? Wq : (wave == 1) ? Wk : (wave == 2) ? Wv : Wg;
  float* O = (wave == 0) ? Qo : (wave == 1) ? Ko : (wave == 2) ? Vo : Go;
  for (int nt = 0; nt < 8; ++nt) {
    v8f acc = {};
#pragma unroll
    for (int kb = 0; kb < DNODE; kb += 4) {
      v2f a  = frag_rm(zs + kb, DNODE, lane);
      v2f bf = frag_rm(W + (size_t)(nt * 16) * DNODE + kb, DNODE, lane);
      acc = wmma4(a, bf, acc);
    }
    const int n  = nt * 16 + (lane & 15);
    const int mo = (lane >> 4) * 8;
#pragma unroll
    for (int r = 0; r < 8; ++r) {
      float val = acc[r];
      if (wave == 3) val = 1.f / (1.f + __expf(-(val + bg[n])));
      O[(size_t)(row0 + mo + r) * DNODE + n] = val;
    }
  }
}

// ------- Kernel 2: fused pair-bias + flash attention (j-split partials) -------
__global__ __launch_bounds__(128) void afp_attn(
    const float* __restrict__ Q, const float* __restrict__ K, const float* __restrict__ V,
    const float* __restrict__ pair, const unsigned char* __restrict__ mask,
    const float* __restrict__ Wb, float* __restrict__ Op, float* __restrict__ Ml)
{
  __shared__ float pairS[2][16 * 16 * DPAIR];   // double-buffered 64KB pair tiles
  __shared__ float wbS[16 * DPAIR];             // Wb zero-padded to 16 "heads"
  __shared__ float biasS[NHEAD * 256];          // [head][i'][j']
  __shared__ float pS[NHEAD * 256];             // per-wave softmax P tile
  const int itile = blockIdx.x & 63;
  const int rest  = blockIdx.x >> 6;
  const int b     = rest & (BATCH - 1);
  const int split = rest >> 1;
  const int i0    = itile * 16;
  const int jbase = split * (LSEQ / JSPLIT);
  const int t = threadIdx.x, wave = t >> 5, lane = t & 31, h = wave;
#pragma unroll
  for (int i = 0; i < 8; ++i) {
    int idx = t + i * 128;
    wbS[idx] = (idx < NHEAD * DPAIR) ? Wb[idx] : 0.f;
  }
  // per-thread slice of the async tile copy: 8 threads per ii row-block,
  // each moves 512B of the contiguous 4KB block pair[b][i0+ii][j0..j0+15][:]
  const int cpy_ii = t >> 3;
  const int cpy_c4 = (t & 7) * 4;               // float offset inside 128B burst
  auto issue = [&](int buf, int j0) {
    const float* gb = pair + (((size_t)(b * LSEQ + i0 + cpy_ii)) * LSEQ + j0) * DPAIR + cpy_c4;
    const float* lb = &pairS[buf][cpy_ii * 1024 + cpy_c4];
    issue_tile_copy(gb, lb);
  };
  issue(0, jbase);
  __syncthreads();                              // wbS ready (async tracked separately)

  const float* Qt = Q + ((size_t)(b * LSEQ + i0)) * DNODE + h * DHEAD;
  v2f qf[8];
#pragma unroll
  for (int kb = 0; kb < 8; ++kb) qf[kb] = frag_rm(Qt + kb * 4, DNODE, lane);

  v8f o0 = {}, o1 = {};
  float mrow[8], lrow[8];
#pragma unroll
  for (int r = 0; r < 8; ++r) { mrow[r] = -__builtin_inff(); lrow[r] = 0.f; }
  const float scale = 0.1767766952966369f;      // Dh^-0.5
  const int n = lane & 15, mo = (lane >> 4) * 8;

  for (int jt = 0; jt < JTILES; ++jt) {
    const int j0 = jbase + jt * 16;
    if (jt + 1 < JTILES) {                      // prefetch next tile, then drain current
      issue((jt + 1) & 1, j0 + 16);
      asm volatile("s_wait_asynccnt 0x20" ::: "memory");
    } else {
      asm volatile("s_wait_asynccnt 0x0" ::: "memory");
    }
    __syncthreads();                            // tile jt resident in LDS for all waves
    const float* pt = pairS[jt & 1];

    // ---- pair bias: [16 j-rows x 64] @ [64 x 16(4 heads)] per i'-row, from LDS
#pragma unroll
    for (int mt = 0; mt < 4; ++mt) {
      const int ii = h * 4 + mt;
      v8f bacc = {};
#pragma unroll
      for (int kb = 0; kb < DPAIR; kb += 4) {
        v2f a = frag_rm(pt + ii * 1024 + kb, DPAIR, lane); // A row = j'
        v2f w = frag_rm(wbS + kb, DPAIR, lane);            // B[k][n] = Wb[n][k]
        bacc = wmma4(a, w, bacc);
      }
      if (n < NHEAD) {
#pragma unroll
        for (int r = 0; r < 8; ++r)
          biasS[n * 256 + ii * 16 + mo + r] = bacc[r];
      }
    }
    __syncthreads();

    // ---- S = Q K^T * scale + bias, with mask
    const float* Kt = K + ((size_t)(b * LSEQ + j0)) * DNODE + h * DHEAD;
    v8f sacc = {};
#pragma unroll
    for (int kb = 0; kb < 8; ++kb) {
      v2f kf = frag_rm(Kt + kb * 4, DNODE, lane);
      sacc = wmma4(qf[kb], kf, sacc);
    }
    const bool mok = mask[(size_t)b * LSEQ + j0 + n] != 0;
    float sv[8];
#pragma unroll
    for (int r = 0; r < 8; ++r) {
      float x = sacc[r] * scale + biasS[h * 256 + (mo + r) * 16 + n];
      sv[r] = mok ? x : -1e9f;
    }
    // ---- online softmax (rows live in 16-lane halves of the wave32)
#pragma unroll
    for (int r = 0; r < 8; ++r) {
      float mx = sv[r];
#pragma unroll
      for (int d = 1; d < 16; d <<= 1) mx = fmaxf(mx, __shfl_xor(mx, d, 32));
      const float mn   = fmaxf(mrow[r], mx);
      const float corr = __expf(mrow[r] - mn);
      mrow[r] = mn;
      const float p = __expf(sv[r] - mn);
      float ps = p;
#pragma unroll
      for (int d = 1; d < 16; d <<= 1) ps += __shfl_xor(ps, d, 32);
      lrow[r] = lrow[r] * corr + ps;
      o0[r] *= corr; o1[r] *= corr;
      pS[h * 256 + (mo + r) * 16 + n] = p;
    }
    asm volatile("s_wait_dscnt 0" ::: "memory");   // same-wave LDS write -> A-frag read
    // ---- O += P @ V
    const float* Vt = V + ((size_t)(b * LSEQ + j0)) * DNODE + h * DHEAD;
#pragma unroll
    for (int kb = 0; kb < 4; ++kb) {
      v2f pf = frag_rm(pS + h * 256 + kb * 4, 16, lane);
      o0 = wmma4(pf, frag_cm(Vt + (size_t)(kb * 4) * DNODE, DNODE, lane), o0);
      o1 = wmma4(pf, frag_cm(Vt + (size_t)(kb * 4) * DNODE + 16, DNODE, lane), o1);
    }
    __syncthreads();   // biasS consumed; pairS[cur] free for the jt+2 copy
  }
  // ---- write per-split partials (unnormalized O, running m, l)
#pragma unroll
  for (int r = 0; r < 8; ++r) {
    const int row = i0 + mo + r;
    const size_t base = (((size_t)split * BATCH + b) * NHEAD + h) * LSEQ + row;
    Op[base * DHEAD + n]      = o0[r];
    Op[base * DHEAD + 16 + n] = o1[r];
    if (n == 0) { Ml[base * 2] = mrow[r]; Ml[base * 2 + 1] = lrow[r]; }
  }
}

// -------- Kernel 3: merge j-split partials, normalize, apply sigmoid gate --------
__global__ __launch_bounds__(256) void afp_merge(
    const float* __restrict__ Op, const float* __restrict__ Ml,
    const float* __restrict__ G, float* __restrict__ XG)
{
  const int idx = blockIdx.x * 256 + threadIdx.x;      // (b*H + h)*L + row
  if (idx >= BATCH * NHEAD * LSEQ) return;
  const int row = idx & (LSEQ - 1);
  const int h   = (idx >> 10) & (NHEAD - 1);
  const int b   = idx >> 12;
  float ms[JSPLIT], ls[JSPLIT], M = -__builtin_inff();
#pragma unroll
  for (int sp = 0; sp < JSPLIT; ++sp) {
    const size_t base = (((size_t)sp * BATCH + b) * NHEAD + h) * LSEQ + row;
    ms[sp] = Ml[base * 2]; ls[sp] = Ml[base * 2 + 1];
    M = fmaxf(M, ms[sp]);
  }
  float w[JSPLIT], l = 0.f;
#pragma unroll
  for (int sp = 0; sp < JSPLIT; ++sp) { w[sp] = __expf(ms[sp] - M); l += w[sp] * ls[sp]; }
  const float invl = 1.f / l;
#pragma unroll
  for (int d = 0; d < DHEAD; ++d) {
    float o = 0.f;
#pragma unroll
    for (int sp = 0; sp < JSPLIT; ++sp)
      o += w[sp] * Op[((((size_t)sp * BATCH + b) * NHEAD + h) * LSEQ + row) * DHEAD + d];
    const size_t gi = ((size_t)(b * LSEQ) + row) * DNODE + h * DHEAD + d;
    XG[gi] = o * invl * G[gi];
  }
}

// ---------------- Kernel 4: out = s + XG @ Wout^T + bout ----------------
__global__ __launch_bounds__(128) void afp_out(
    const float* __restrict__ XG, const float* __restrict__ s,
    const float* __restrict__ Wout, const float* __restrict__ bout,
    float* __restrict__ out)
{
  __shared__ float xs[16 * DNODE];
  const int row0 = blockIdx.x * 16;
  const int t = threadIdx.x;
#pragma unroll
  for (int i = 0; i < 16; ++i)
    xs[i * 128 + t] = XG[(size_t)row0 * DNODE + i * 128 + t];
  __syncthreads();
  const int wave = t >> 5, lane = t & 31;
  for (int half = 0; half < 2; ++half) {
    const int nt = wave * 2 + half;
    v8f acc = {};
#pragma unroll
    for (int kb = 0; kb < DNODE; kb += 4) {
      v2f a = frag_rm(xs + kb, DNODE, lane);
      v2f w = frag_rm(Wout + (size_t)(nt * 16) * DNODE + kb, DNODE, lane);
      acc = wmma4(a, w, acc);
    }
    const int n  = nt * 16 + (lane & 15);
    const int mo = (lane >> 4) * 8;
#pragma unroll
    for (int r = 0; r < 8; ++r) {
      const size_t idx = (size_t)(row0 + mo + r) * DNODE + n;
      out[idx] = s[idx] + acc[r] + bout[n];
    }
  }
}

extern "C" void kernel_launch(void* const* d_in, const int* in_sizes, int n_in,
                              void* d_out, int out_size, void* d_ws, size_t ws_size,
                              hipStream_t stream) {
  (void)in_sizes; (void)n_in; (void)out_size; (void)ws_size;
  const float* s    = (const float*)d_in[0];
  const float* pair = (const float*)d_in[1];
  const unsigned char* mask = (const unsigned char*)d_in[2];
  const float* lnw  = (const float*)d_in[3];
  const float* lnb  = (const float*)d_in[4];
  const float* Wq   = (const float*)d_in[5];
  const float* Wk   = (const float*)d_in[6];
  const float* Wv   = (const float*)d_in[7];
  const float* Wb   = (const float*)d_in[8];
  const float* Wg   = (const float*)d_in[9];
  const float* bg   = (const float*)d_in[10];
  const float* Wout = (const float*)d_in[11];
  const float* bout = (const float*)d_in[12];
  float* out = (float*)d_out;
  float* ws  = (float*)d_ws;
  const size_t nrow = (size_t)BATCH * LSEQ;     // 2048
  float* Qw = ws;
  float* Kw = Qw + nrow * DNODE;
  float* Vw = Kw + nrow * DNODE;
  float* Gw = Vw + nrow * DNODE;
  float* XG = Gw + nrow * DNODE;
  float* Op = XG + nrow * DNODE;                             // JSPLIT*B*H*L*32 = 4MB
  float* Ml = Op + (size_t)JSPLIT * BATCH * NHEAD * LSEQ * DHEAD;  // + 256KB
  const int blocks = (int)(nrow / 16);          // 128
  afp_proj<<<blocks, 128, 0, stream>>>(s, lnw, lnb, Wq, Wk, Wv, Wg, bg, Qw, Kw, Vw, Gw);
  afp_attn<<<JSPLIT * BATCH * (LSEQ / 16), 128, 0, stream>>>(Qw, Kw, Vw, pair, mask, Wb, Op, Ml);
  afp_merge<<<(BATCH * NHEAD * LSEQ + 255) / 256, 256, 0, stream>>>(Op, Ml, Gw, XG);
  afp_out<<<blocks, 128, 0, stream>>>(XG, s, Wout, bout, out);
}